// GraphSAGE_48198122995902
// MI455X (gfx1250) — compile-verified
//
#include <hip/hip_runtime.h>
#include <hip/hip_bf16.h>

typedef __attribute__((ext_vector_type(2))) float v2f;
typedef __attribute__((ext_vector_type(8))) float v8f;

#define N_NODES 50000
#define N_EDGES 300000
#define D 256
#define K_TOT 512
#define LDA 516  // padded LDS row stride (floats): 4*row mod 64 -> conflict-free

// workspace layout (float offsets)
static const size_t OFF_SUM_OBJ = 0;
static const size_t OFF_SUM_REL = (size_t)N_NODES * D;
static const size_t OFF_CNT     = OFF_SUM_REL + (size_t)N_NODES * D;
static const size_t OFF_BT_OBJ  = OFF_CNT + N_NODES;             // even
static const size_t OFF_BT_REL  = OFF_BT_OBJ + (size_t)K_TOT * D;

__global__ void zero_kernel(float* __restrict__ p, size_t n) {
    size_t i = (size_t)blockIdx.x * blockDim.x + threadIdx.x;
    if (i < n) p[i] = 0.0f;
}

// Build combined B in k-major float2-pair layout:
// bt[(k/2)*D + j] = { B[k][j], B[k+1][j] },  B[k][j] = k<256 ? W_l[j,k] : W_r[j,k-256]
__global__ void prep_b_kernel(const float* __restrict__ Wl_o, const float* __restrict__ Wr_o,
                              const float* __restrict__ Wl_r, const float* __restrict__ Wr_r,
                              v2f* __restrict__ bt_o, v2f* __restrict__ bt_r) {
    int idx = blockIdx.x * blockDim.x + threadIdx.x;
    if (idx >= 2 * (K_TOT / 2) * D) return;
    int conv = idx >> 16;          // 65536 pairs per conv
    int rem  = idx & 0xFFFF;
    int kp   = rem >> 8;           // k-pair index, k = 2*kp (both k in same half)
    int j    = rem & (D - 1);
    const float* Wl = conv ? Wl_r : Wl_o;
    const float* Wr = conv ? Wr_r : Wr_o;
    int k0 = kp * 2;
    v2f v;
    if (k0 < D) {
        v.x = Wl[j * D + k0];
        v.y = Wl[j * D + k0 + 1];
    } else {
        v.x = Wr[j * D + (k0 - D)];
        v.y = Wr[j * D + (k0 - D) + 1];
    }
    (conv ? bt_r : bt_o)[kp * D + j] = v;
}

__global__ void count_kernel(const long long* __restrict__ edge, float* __restrict__ cnt, int e_cnt) {
    int e = blockIdx.x * blockDim.x + threadIdx.x;
    if (e < e_cnt) {
        int dst = (int)edge[2 * e + 1];
        unsafeAtomicAdd(&cnt[dst], 1.0f);
    }
}

// one block per edge, 64 threads, float4 per thread, per-float f32 atomics
__global__ __launch_bounds__(64) void scatter_kernel(const long long* __restrict__ edge,
                                                     const float* __restrict__ x_obj,
                                                     const float* __restrict__ x_rel,
                                                     float* __restrict__ sum_obj,
                                                     float* __restrict__ sum_rel) {
    int e = blockIdx.x;
    int t = threadIdx.x;                 // 0..63 -> features 4t..4t+3
    long long s = edge[2 * e];
    long long d = edge[2 * e + 1];
    float4 vo = ((const float4*)(x_obj + (size_t)s * D))[t];
    float4 vr = ((const float4*)(x_rel + (size_t)s * D))[t];
    float* qo = sum_obj + (size_t)d * D + 4 * t;
    float* qr = sum_rel + (size_t)d * D + 4 * t;
    unsafeAtomicAdd(qo + 0, vo.x);
    unsafeAtomicAdd(qo + 1, vo.y);
    unsafeAtomicAdd(qo + 2, vo.z);
    unsafeAtomicAdd(qo + 3, vo.w);
    unsafeAtomicAdd(qr + 0, vr.x);
    unsafeAtomicAdd(qr + 1, vr.y);
    unsafeAtomicAdd(qr + 2, vr.z);
    unsafeAtomicAdd(qr + 3, vr.w);
}

// Fused SAGE output GEMM: out = [agg | x] (Nx512) @ Bt (512x256) + bias
// block: 512 threads = 16 waves; block = 16 M-rows, wave = 16 N-cols; K via WMMA f32 16x16x4
__global__ __launch_bounds__(512) void sage_gemm_kernel(
    const float* __restrict__ x_obj, const float* __restrict__ x_rel,
    const float* __restrict__ sum_obj, const float* __restrict__ sum_rel,
    const float* __restrict__ cnt,
    const v2f* __restrict__ bt_obj, const v2f* __restrict__ bt_rel,
    const float* __restrict__ b_obj, const float* __restrict__ b_rel,
    float* __restrict__ out) {
    __shared__ float sA[16 * LDA];

    const int conv = blockIdx.y;
    const int m0   = blockIdx.x * 16;
    const float* x   = conv ? x_rel   : x_obj;
    const float* sum = conv ? sum_rel : sum_obj;
    const v2f*   bt  = conv ? bt_rel  : bt_obj;
    const float* bv  = conv ? b_rel   : b_obj;
    float* outp = out + (size_t)conv * N_NODES * D;

    // stage A = [summed/deg | x] tile (16 x 512) into LDS
    {
        int t   = threadIdx.x;       // 0..511
        int row = t >> 5;            // 0..15
        int c0  = (t & 31) * 16;     // 0..496, 16 floats per thread, one half only
        int g   = m0 + row;          // N divisible by 16 -> always in range
        if (c0 < D) {
            float inv = 1.0f / fmaxf(cnt[g], 1.0f);
            const float4* src = (const float4*)(sum + (size_t)g * D + c0);
#pragma unroll
            for (int i = 0; i < 4; ++i) {
                float4 v = src[i];
                sA[row * LDA + c0 + 4 * i + 0] = v.x * inv;
                sA[row * LDA + c0 + 4 * i + 1] = v.y * inv;
                sA[row * LDA + c0 + 4 * i + 2] = v.z * inv;
                sA[row * LDA + c0 + 4 * i + 3] = v.w * inv;
            }
        } else {
            const float4* src = (const float4*)(x + (size_t)g * D + (c0 - D));
#pragma unroll
            for (int i = 0; i < 4; ++i) {
                float4 v = src[i];
                sA[row * LDA + c0 + 4 * i + 0] = v.x;
                sA[row * LDA + c0 + 4 * i + 1] = v.y;
                sA[row * LDA + c0 + 4 * i + 2] = v.z;
                sA[row * LDA + c0 + 4 * i + 3] = v.w;
            }
        }
    }
    __syncthreads();

    const int wave  = threadIdx.x >> 5;   // 0..15 -> N-tile
    const int lane  = threadIdx.x & 31;
    const int mrow  = lane & 15;          // A fragment row (lanes 0-15 / 16-31 both M=0..15)
    const int khalf = lane >> 4;          // 0: K=k..k+1, 1: K=k+2..k+3
    const int jj    = wave * 16 + mrow;   // output column (B/C col = lane&15)

    v8f acc;
    {
        float bias = bv[jj];
#pragma unroll
        for (int r = 0; r < 8; ++r) acc[r] = bias;
    }

#pragma unroll 8
    for (int k = 0; k < K_TOT; k += 4) {
        int kk = k + 2 * khalf;
        v2f a = *(const v2f*)(&sA[mrow * LDA + kk]);     // ds_load_b64, bank-conflict-free
        v2f b = bt[(size_t)(kk >> 1) * D + jj];          // global_load_b64, coalesced, L2-resident
        acc = __builtin_amdgcn_wmma_f32_16x16x4_f32(
            /*neg_a=*/false, a, /*neg_b=*/false, b,
            /*c_mod=*/(short)0, acc, /*reuse_a=*/false, /*reuse_b=*/false);
    }

    // C/D layout: VGPR r -> M = r + 8*(lane>=16), N = lane&15
#pragma unroll
    for (int r = 0; r < 8; ++r) {
        int grow = m0 + r + 8 * khalf;
        outp[(size_t)grow * D + jj] = acc[r];
    }
}

extern "C" void kernel_launch(void* const* d_in, const int* in_sizes, int n_in,
                              void* d_out, int out_size, void* d_ws, size_t ws_size,
                              hipStream_t stream) {
    const float*     obj_vecs = (const float*)d_in[0];
    const float*     rel_vecs = (const float*)d_in[1];
    const long long* edge     = (const long long*)d_in[2];
    const float*     W_l_obj  = (const float*)d_in[3];
    const float*     b_l_obj  = (const float*)d_in[4];
    const float*     W_r_obj  = (const float*)d_in[5];
    const float*     W_l_rel  = (const float*)d_in[6];
    const float*     b_l_rel  = (const float*)d_in[7];
    const float*     W_r_rel  = (const float*)d_in[8];

    float* ws      = (float*)d_ws;
    float* sum_obj = ws + OFF_SUM_OBJ;
    float* sum_rel = ws + OFF_SUM_REL;
    float* cnt     = ws + OFF_CNT;
    v2f*   bt_obj  = (v2f*)(ws + OFF_BT_OBJ);
    v2f*   bt_rel  = (v2f*)(ws + OFF_BT_REL);

    // 1) zero accumulators + counts (harness does not re-poison between replays)
    size_t zn = OFF_BT_OBJ;  // summed_obj + summed_rel + cnt
    zero_kernel<<<(unsigned)((zn + 255) / 256), 256, 0, stream>>>(ws, zn);

    // 2) transpose/pack weights into k-major pair layout
    int nprep = 2 * (K_TOT / 2) * D;
    prep_b_kernel<<<(nprep + 255) / 256, 256, 0, stream>>>(
        W_l_obj, W_r_obj, W_l_rel, W_r_rel, bt_obj, bt_rel);

    // 3) degree counts
    count_kernel<<<(N_EDGES + 255) / 256, 256, 0, stream>>>(edge, cnt, N_EDGES);

    // 4) gather + scatter-add messages (both convs share edge_index)
    scatter_kernel<<<N_EDGES, 64, 0, stream>>>(edge, obj_vecs, rel_vecs, sum_obj, sum_rel);

    // 5) fused GEMM epilogue via f32 WMMA (grid.y = conv)
    dim3 g(N_NODES / 16, 2);
    sage_gemm_kernel<<<g, 512, 0, stream>>>(
        obj_vecs, rel_vecs, sum_obj, sum_rel, cnt,
        bt_obj, bt_rel, b_l_obj, b_l_rel, (float*)d_out);
}